// GIN_node_classification_74431783240459
// MI455X (gfx1250) — compile-verified
//
#include <hip/hip_runtime.h>
#include <hip/hip_bf16.h>

#define N_NODES   50000
#define IN_F      128
#define HID       256
#define N_CLASSES 2
#define N_EDGES   640000
#define BN_EPS    1e-5f

typedef __attribute__((ext_vector_type(2))) float v2f;
typedef __attribute__((ext_vector_type(8))) float v8f;

// ---------------------------------------------------------------------------
// Kernel 1: BN0 (eval-mode affine) -> HB, and init AGG = (1+eps)*h = h
// ---------------------------------------------------------------------------
__global__ void bn0_init_kernel(const float* __restrict__ x,
                                const float* __restrict__ g0,
                                const float* __restrict__ b0,
                                const float* __restrict__ rm0,
                                const float* __restrict__ rv0,
                                float* __restrict__ HB,
                                float* __restrict__ AGG) {
    int idx = blockIdx.x * blockDim.x + threadIdx.x;   // exact grid: N*F/256
    int f = idx & (IN_F - 1);
    float s = g0[f] * rsqrtf(rv0[f] + BN_EPS);
    float v = (x[idx] - rm0[f]) * s + b0[f];
    HB[idx]  = v;
    AGG[idx] = v;
}

// ---------------------------------------------------------------------------
// Kernel 2: fold BN1 into W2/b2:  W2'[k][c] = s1[k]*W2[k][c]
//           b2'[c] = b2[c] + sum_k (beta1[k] - rm1[k]*s1[k]) * W2[k][c]
// ---------------------------------------------------------------------------
__global__ void fold_w2_kernel(const float* __restrict__ g1,
                               const float* __restrict__ b1,
                               const float* __restrict__ rm1,
                               const float* __restrict__ rv1,
                               const float* __restrict__ W2,
                               const float* __restrict__ bias2,
                               float* __restrict__ W2P,
                               float* __restrict__ B2P) {
    __shared__ float acc0[HID];
    __shared__ float acc1[HID];
    int k = threadIdx.x;                 // one block of HID threads
    float s  = g1[k] * rsqrtf(rv1[k] + BN_EPS);
    float sh = b1[k] - rm1[k] * s;
    float w0 = W2[k * 2 + 0];
    float w1 = W2[k * 2 + 1];
    W2P[k * 2 + 0] = s * w0;
    W2P[k * 2 + 1] = s * w1;
    acc0[k] = sh * w0;
    acc1[k] = sh * w1;
    __syncthreads();
    for (int st = HID / 2; st > 0; st >>= 1) {
        if (k < st) { acc0[k] += acc0[k + st]; acc1[k] += acc1[k + st]; }
        __syncthreads();
    }
    if (k == 0) { B2P[0] = bias2[0] + acc0[0]; B2P[1] = bias2[1] + acc1[0]; }
}

// ---------------------------------------------------------------------------
// Kernel 3: GIN scatter-sum. 32 lanes per edge, 4 features per lane.
//           AGG[dst] += HB[src]   (atomic fp32, stays in 192MB L2)
// ---------------------------------------------------------------------------
__global__ void scatter_kernel(const float* __restrict__ HB,
                               const long long* __restrict__ ei,
                               float* __restrict__ AGG) {
    unsigned gid = blockIdx.x * blockDim.x + threadIdx.x; // exact: E*32
    unsigned lane = gid & 31u;
    unsigned e    = gid >> 5;
    int src = (int)ei[e];
    int dst = (int)ei[e + N_EDGES];
    const float4 v = *(const float4*)(HB + (size_t)src * IN_F + lane * 4);
    float* base = AGG + (size_t)dst * IN_F + lane * 4;
    atomicAdd(base + 0, v.x);
    atomicAdd(base + 1, v.y);
    atomicAdd(base + 2, v.z);
    atomicAdd(base + 3, v.w);
}

// ---------------------------------------------------------------------------
// Kernel 4: H1 = relu(AGG @ W1 + bias1) via V_WMMA_F32_16X16X4_F32.
// One wave per 16x16 tile. Grid: 6250 blocks x 256 threads (8 waves).
// Block b: mt = b>>1 (uniform), nt = (b&1)*8 + waveId. M=50000=16*3125 exact,
// so EXEC is all-1s for every WMMA (ISA requirement).
//
// Fragment layouts (ISA 7.12.2, 32-bit wave32):
//   A 16x4:  lane -> row M = lane%16 ; VGPR j -> K = 2*(lane/16) + j
//   B 4x16:  lane -> col N = lane%16 ; VGPR j -> K = 2*(lane/16) + j
//   C 16x16: lane -> col N = lane%16 ; VGPR r -> row M = r + 8*(lane/16)
// ---------------------------------------------------------------------------
__global__ void gemm1_wmma_kernel(const float* __restrict__ AGG,
                                  const float* __restrict__ W1,
                                  const float* __restrict__ bias1,
                                  float* __restrict__ H1) {
    __shared__ float As[16][132];          // 16 rows x 128 K, padded stride 132

    const int mt     = blockIdx.x >> 1;            // 0..3124
    const int ntbase = (blockIdx.x & 1) * 8;       // 0 or 8
    const int waveId = threadIdx.x >> 5;           // 0..7
    const int nt     = ntbase + waveId;            // 0..15
    const int lane   = threadIdx.x & 31;
    const int hl     = lane >> 4;                  // half-wave: 0 or 1
    const int ll     = lane & 15;

    // --- stage the 16x128 A-tile (AGG rows) into LDS: 8 floats/thread ---
    {
        const int row = threadIdx.x >> 4;          // 0..15
        const int k   = (threadIdx.x & 15) * 8;    // 0..120
        const float* sp = AGG + (size_t)(mt * 16 + row) * IN_F + k;
        float4 p0 = *(const float4*)(sp);
        float4 p1 = *(const float4*)(sp + 4);
        As[row][k + 0] = p0.x; As[row][k + 1] = p0.y;
        As[row][k + 2] = p0.z; As[row][k + 3] = p0.w;
        As[row][k + 4] = p1.x; As[row][k + 5] = p1.y;
        As[row][k + 6] = p1.z; As[row][k + 7] = p1.w;
    }
    __syncthreads();

    const int n = nt * 16 + ll;                    // output column of this lane
    v8f c = {};

#pragma unroll 4
    for (int k0 = 0; k0 < IN_F; k0 += 4) {
        const int ka = k0 + 2 * hl;                // K index of fragment elem 0
        v2f a, b;
        a.x = As[ll][ka];
        a.y = As[ll][ka + 1];
        b.x = W1[(size_t)(ka + 0) * HID + n];
        b.y = W1[(size_t)(ka + 1) * HID + n];
        c = __builtin_amdgcn_wmma_f32_16x16x4_f32(
                /*neg_a=*/false, a, /*neg_b=*/false, b,
                /*c_mod=*/(short)0, c, /*reuse_a=*/false, /*reuse_b=*/false);
    }

    // --- epilogue: +bias1, ReLU, store to H1[50000][256] ---
    const float bn = bias1[n];
#pragma unroll
    for (int r = 0; r < 8; ++r) {
        const int mrow = mt * 16 + r + 8 * hl;
        float z = c[r] + bn;
        H1[(size_t)mrow * HID + n] = z > 0.f ? z : 0.f;
    }
}

// ---------------------------------------------------------------------------
// Kernel 5: out = H1 @ W2' + b2'   (K=256, N=2: too skinny for WMMA)
// ---------------------------------------------------------------------------
__global__ void out_kernel(const float* __restrict__ H1,
                           const float* __restrict__ W2P,
                           const float* __restrict__ B2P,
                           float* __restrict__ out) {
    int i = blockIdx.x * blockDim.x + threadIdx.x;
    if (i >= N_NODES) return;
    const float* h = H1 + (size_t)i * HID;
    float a0 = 0.f, a1 = 0.f;
#pragma unroll 8
    for (int k = 0; k < HID; k += 4) {
        float4 hv = *(const float4*)(h + k);
        a0 = fmaf(hv.x, W2P[(k + 0) * 2 + 0], a0);
        a1 = fmaf(hv.x, W2P[(k + 0) * 2 + 1], a1);
        a0 = fmaf(hv.y, W2P[(k + 1) * 2 + 0], a0);
        a1 = fmaf(hv.y, W2P[(k + 1) * 2 + 1], a1);
        a0 = fmaf(hv.z, W2P[(k + 2) * 2 + 0], a0);
        a1 = fmaf(hv.z, W2P[(k + 2) * 2 + 1], a1);
        a0 = fmaf(hv.w, W2P[(k + 3) * 2 + 0], a0);
        a1 = fmaf(hv.w, W2P[(k + 3) * 2 + 1], a1);
    }
    out[i * 2 + 0] = a0 + B2P[0];
    out[i * 2 + 1] = a1 + B2P[1];
}

// ---------------------------------------------------------------------------
extern "C" void kernel_launch(void* const* d_in, const int* in_sizes, int n_in,
                              void* d_out, int out_size, void* d_ws, size_t ws_size,
                              hipStream_t stream) {
    // setup_inputs() order:
    const float*     x      = (const float*)d_in[0];
    const long long* ei     = (const long long*)d_in[1];
    const float*     gamma0 = (const float*)d_in[2];
    const float*     beta0  = (const float*)d_in[3];
    const float*     rm0    = (const float*)d_in[4];
    const float*     rv0    = (const float*)d_in[5];
    const float*     W1     = (const float*)d_in[6];
    const float*     bias1  = (const float*)d_in[7];
    const float*     gamma1 = (const float*)d_in[8];
    const float*     beta1  = (const float*)d_in[9];
    const float*     rm1    = (const float*)d_in[10];
    const float*     rv1    = (const float*)d_in[11];
    const float*     W2     = (const float*)d_in[12];
    const float*     bias2  = (const float*)d_in[13];
    float* out = (float*)d_out;

    // Workspace layout
    char* ws = (char*)d_ws;
    const size_t SZ_HB = (size_t)N_NODES * IN_F * sizeof(float);   // 25.6 MB
    const size_t SZ_H1 = (size_t)N_NODES * HID * sizeof(float);    // 51.2 MB
    float* HB  = (float*)(ws);
    float* AGG = (float*)(ws + SZ_HB);
    float* H1  = (float*)(ws + 2 * SZ_HB);
    float* W2P = (float*)(ws + 2 * SZ_HB + SZ_H1);
    float* B2P = W2P + HID * N_CLASSES;

    // 1) BN0 + AGG init: 6,400,000 elems / 256 = 25000 blocks (exact)
    bn0_init_kernel<<<(N_NODES * IN_F) / 256, 256, 0, stream>>>(
        x, gamma0, beta0, rm0, rv0, HB, AGG);

    // 2) Fold BN1 into W2/b2
    fold_w2_kernel<<<1, HID, 0, stream>>>(
        gamma1, beta1, rm1, rv1, W2, bias2, W2P, B2P);

    // 3) Edge scatter-sum: 640000 * 32 lanes / 256 = 80000 blocks (exact)
    scatter_kernel<<<(N_EDGES * 32) / 256, 256, 0, stream>>>(HB, ei, AGG);

    // 4) WMMA GEMM1 + bias + ReLU: 3125 M-tiles * 16 N-tiles / 8 waves = 6250
    gemm1_wmma_kernel<<<6250, 256, 0, stream>>>(AGG, W1, bias1, H1);

    // 5) Output linear (BN1 folded): one thread per node
    out_kernel<<<(N_NODES + 255) / 256, 256, 0, stream>>>(H1, W2P, B2P, out);
}